// LoRALinear_44152263803472
// MI455X (gfx1250) — compile-verified
//
#include <hip/hip_runtime.h>

typedef __attribute__((ext_vector_type(2))) float v2f;
typedef __attribute__((ext_vector_type(4))) float v4f;
typedef __attribute__((ext_vector_type(8))) float v8f;
typedef __attribute__((ext_vector_type(4))) int   i4;

#define IN_F   4096
#define OUT_F  4096
#define RANK   8
#define LSCALE 1.0f   // alpha / rank = 8 / 8

#define BM  128
#define BN  128
#define BK  32
#define PAD 36        // BK + 4 floats: bank-conflict-free LDS rows (gcd(36,64)=4)

#define AS1 __attribute__((address_space(1)))
#define AS3 __attribute__((address_space(3)))

// ---- CDNA5 async global->LDS path (probe via __has_builtin, sync fallback) ----
#if __has_builtin(__builtin_amdgcn_global_load_async_to_lds_b128) && \
    __has_builtin(__builtin_amdgcn_global_load_async_to_lds_b32)
#define HAVE_ASYNC_LDS 1
#else
#define HAVE_ASYNC_LDS 0
#endif

__device__ __forceinline__ void cp_b128(const float* g, float* l) {
#if HAVE_ASYNC_LDS
    __builtin_amdgcn_global_load_async_to_lds_b128(
        (AS1 i4*)(i4*)const_cast<float*>(g), (AS3 i4*)(i4*)l, 0, 0);
#else
    *(v4f*)l = *(const v4f*)g;
#endif
}
__device__ __forceinline__ void cp_b32(const float* g, float* l) {
#if HAVE_ASYNC_LDS
    __builtin_amdgcn_global_load_async_to_lds_b32(
        (AS1 int*)(int*)const_cast<float*>(g), (AS3 int*)(int*)l, 0, 0);
#else
    *l = *g;
#endif
}

#if HAVE_ASYNC_LDS
#  if __has_builtin(__builtin_amdgcn_s_wait_asynccnt)
#    define WAIT_ASYNC(n) __builtin_amdgcn_s_wait_asynccnt(n)
#  else
#    define WAIT_ASYNC(n) asm volatile("s_wait_asynccnt %0" ::"n"(n) : "memory")
#  endif
#else
#  define WAIT_ASYNC(n) ((void)0)
#endif

// out[m][n] = sum_k X[m][k]*W[n][k] + b[n] + LSCALE * sum_r (X@A^T)[m][r] * B[n][r]
__global__ __launch_bounds__(256)
void lora_wmma_f32_kernel(const float* __restrict__ X,
                          const float* __restrict__ W,
                          const float* __restrict__ bias,
                          const float* __restrict__ lA,
                          const float* __restrict__ lB,
                          float* __restrict__ Out)
{
    __shared__ float Xs[2][BM * PAD];     // 2 x 18432 B   X tile  [m][k]
    __shared__ float Ws[2][BN * PAD];     // 2 x 18432 B   W tile  [n][k]
    __shared__ float As[2][RANK * PAD];   // 2 x  1152 B   lora_A  [r][k]
    __shared__ float xaS[BM * RANK];      //      4096 B   X @ A^T [m][r]
    __shared__ float blS[BN * RANK];      //      4096 B   lora_B  [n][r]
    __shared__ float biasS[BN];           //       512 B

    const int tid  = threadIdx.x;
    const int wave = tid >> 5;            // 0..7
    const int lane = tid & 31;
    const int half = lane >> 4;           // 0 | 1
    const int l16  = lane & 15;

    const int wm = wave >> 2;             // 0..1  (64-row slice of M)
    const int wn = wave & 3;              // 0..3  (32-col slice of N)

    const long m0 = (long)blockIdx.y * BM;
    const long n0 = (long)blockIdx.x * BN;

    // global->LDS tile mapping: 128x32 floats, 256 threads, 4 passes of b128
    const int gcol4 = tid & 7;            // b128 column within BK
    const int grow  = tid >> 3;           // 0..31 (+32 per pass)

    // 9 async ops per wave per stage: 4 (X) + 4 (W) + 1 (lora_A), all lanes active
    auto stage_tiles = [&](int s, int k0) {
#pragma unroll
        for (int pass = 0; pass < 4; ++pass) {
            const int row = grow + pass * 32;
            cp_b128(X + (m0 + row) * (long)IN_F + k0 + gcol4 * 4,
                    &Xs[s][row * PAD + gcol4 * 4]);
            cp_b128(W + (n0 + row) * (long)IN_F + k0 + gcol4 * 4,
                    &Ws[s][row * PAD + gcol4 * 4]);
        }
        cp_b32(lA + wave * (long)IN_F + k0 + lane, &As[s][wave * PAD + lane]);
    };

    // ---- one-time tile preloads: bias and lora_B ----
    if (tid < BN) biasS[tid] = bias[n0 + tid];
    {
        const int r = tid >> 1, p = tid & 1;              // 256 float4, 1/thread
        *(v4f*)(&blS[r * RANK + p * 4]) = *(const v4f*)(lB + (n0 + r) * RANK + p * 4);
    }

    v8f acc[4][2];
#pragma unroll
    for (int mi = 0; mi < 4; ++mi)
#pragma unroll
        for (int ni = 0; ni < 2; ++ni)
            acc[mi][ni] = v8f{0, 0, 0, 0, 0, 0, 0, 0};

    // LoRA partials: lane covers (row = wave*16 + l16, r = half*4 + j)
    float xa[4] = {0.f, 0.f, 0.f, 0.f};
    const int xaRow = wave * 16 + l16;
    const int rbase = half * 4;

    stage_tiles(0, 0);                                    // prologue

    int s = 0;
    for (int k0 = 0; k0 < IN_F; k0 += BK, s ^= 1) {
        if (k0 + BK < IN_F) {
            stage_tiles(s ^ 1, k0 + BK);                  // prefetch next stage
            WAIT_ASYNC(9);                                // stage s retired (in-order)
        } else {
            WAIT_ASYNC(0);
        }
        __syncthreads();                                  // all waves' stage s visible

        const float* __restrict__ Xt = &Xs[s][0];
        const float* __restrict__ Wt = &Ws[s][0];
        const float* __restrict__ At = &As[s][0];

        // ---- dense GEMM: 8 k-steps of V_WMMA_F32_16X16X4_F32 ----
#pragma unroll
        for (int kk = 0; kk < BK; kk += 4) {
            v2f a_frag[4], b_frag[2];
#pragma unroll
            for (int mi = 0; mi < 4; ++mi) {
                const int row = wm * 64 + mi * 16 + l16;
                a_frag[mi] = *(const v2f*)(&Xt[row * PAD + kk + half * 2]);
            }
#pragma unroll
            for (int ni = 0; ni < 2; ++ni) {
                const int row = wn * 32 + ni * 16 + l16;
                b_frag[ni] = *(const v2f*)(&Wt[row * PAD + kk + half * 2]);
            }
#pragma unroll
            for (int mi = 0; mi < 4; ++mi)
#pragma unroll
                for (int ni = 0; ni < 2; ++ni)
                    acc[mi][ni] = __builtin_amdgcn_wmma_f32_16x16x4_f32(
                        false, a_frag[mi], false, b_frag[ni],
                        (short)0, acc[mi][ni], false, false);
        }

        // ---- fused LoRA rank-8 accumulation from the same LDS X tile ----
#pragma unroll 4
        for (int k = 0; k < BK; ++k) {
            const float xv = Xt[xaRow * PAD + k];
#pragma unroll
            for (int j = 0; j < 4; ++j)
                xa[j] += xv * At[(rbase + j) * PAD + k];
        }
        __syncthreads();                                  // stage s free to overwrite
    }

    // publish xa = X @ A^T for this M tile
#pragma unroll
    for (int j = 0; j < 4; ++j)
        xaS[xaRow * RANK + rbase + j] = xa[j];
    __syncthreads();

    // ---- epilogue: out = acc + bias + LSCALE * xa . loraB ----
    v4f  blv[2][2];
    float bv[2];
#pragma unroll
    for (int ni = 0; ni < 2; ++ni) {
        const int col = wn * 32 + ni * 16 + l16;
        blv[ni][0] = *(const v4f*)(&blS[col * RANK]);
        blv[ni][1] = *(const v4f*)(&blS[col * RANK + 4]);
        bv[ni]     = biasS[col];
    }

#pragma unroll
    for (int mi = 0; mi < 4; ++mi) {
#pragma unroll
        for (int j = 0; j < 8; ++j) {
            // C/D layout: VGPR j -> M = j (lanes 0-15) | j+8 (lanes 16-31)
            const int row = wm * 64 + mi * 16 + j + half * 8;
            v4f xlo = *(const v4f*)(&xaS[row * RANK]);
            v4f xhi = *(const v4f*)(&xaS[row * RANK + 4]);
#pragma unroll
            for (int ni = 0; ni < 2; ++ni) {
                float lor = 0.f;
#pragma unroll
                for (int r = 0; r < 4; ++r)
                    lor += xlo[r] * blv[ni][0][r] + xhi[r] * blv[ni][1][r];
                const int col = wn * 32 + ni * 16 + l16;
                const float val = acc[mi][ni][j] + bv[ni] + LSCALE * lor;
                Out[(m0 + row) * (long)OUT_F + (n0 + col)] = val;
            }
        }
    }
}

extern "C" void kernel_launch(void* const* d_in, const int* in_sizes, int n_in,
                              void* d_out, int out_size, void* d_ws, size_t ws_size,
                              hipStream_t stream) {
    const float* X    = (const float*)d_in[0];   // [B*S, 4096]
    const float* W    = (const float*)d_in[1];   // [4096, 4096]
    const float* bias = (const float*)d_in[2];   // [4096]
    const float* lA   = (const float*)d_in[3];   // [8, 4096]
    const float* lB   = (const float*)d_in[4];   // [4096, 8]
    float* Out = (float*)d_out;

    const int Mtot = in_sizes[0] / IN_F;         // 16384
    dim3 grid(OUT_F / BN, Mtot / BM);            // N fastest -> X tile + W shared in L2
    dim3 block(256);
    lora_wmma_f32_kernel<<<grid, block, 0, stream>>>(X, W, bias, lA, lB, Out);
}